// FusionNet_48558900248753
// MI455X (gfx1250) — compile-verified
//
#include <hip/hip_runtime.h>
#include <math.h>

// ---------------------------------------------------------------------------
// MI455X (gfx1250) implementation of the fused spectral pipeline.
// All DCT/FFT transforms are expressed as 256x256 fp32 GEMMs executed with
// V_WMMA_F32_16X16X4_F32 (wave32 WMMA). SVD truncation is done via a Gram
// matrix (split-K WMMA GEMM) + parallel Jacobi eigensolver per batch.
// k_gemm256: compile-time strides -> immediate-offset loads, no address VALU.
// ---------------------------------------------------------------------------

typedef float v2f __attribute__((ext_vector_type(2)));
typedef float v8f __attribute__((ext_vector_type(8)));

#define NIMG 512            // B*C
#define HW   65536          // 256*256
#define NPIX 256
#define CCH  64
#define BB   8

__device__ __forceinline__ v8f wmma4(v2f a, v2f b, v8f c) {
  // D = A(16x4) x B(4x16) + C(16x16), fp32
  return __builtin_amdgcn_wmma_f32_16x16x4_f32(
      false, a, false, b, (short)0, c, false, false);
}

// ---------------------------------------------------------------------------
// Specialized GEMM: C = alpha*A*B (+C), M=N=K=256, lda=ldb=ldc=256.
// block = 256 threads = 8 waves arranged 2(M) x 4(N); each wave -> 32x32 tile
// (4 accumulators). Full K unroll: every load gets an immediate offset.
// grid = (2, 4, nimg)
// ---------------------------------------------------------------------------
__global__ void k_gemm256(const float* __restrict__ A, long long sA,
                          const float* __restrict__ B, long long sB,
                          float* __restrict__ C, long long sC,
                          float alphaC, const float* aP0, const float* aP1,
                          int accumulate)
{
  const int img = blockIdx.z;
  const float* Ai = A + (size_t)img * (size_t)sA;
  const float* Bi = B + (size_t)img * (size_t)sB;
  float*       Ci = C + (size_t)img * (size_t)sC;

  const int wave = threadIdx.x >> 5;
  const int lane = threadIdx.x & 31;
  const int wm = wave >> 2, wn = wave & 3;        // 2 x 4 waves
  const int tm = blockIdx.y * 64  + wm * 32;
  const int tn = blockIdx.x * 128 + wn * 32;
  const int r  = lane & 15;
  const int hi = lane >> 4;
  const int kb0 = hi ? 2 : 0;

  const float* A0 = Ai + (size_t)(tm + r) * 256 + kb0;   // rows tm..tm+15
  const float* A1 = A0 + 16 * 256;                       // rows tm+16..tm+31
  const float* B0 = Bi + (size_t)kb0 * 256 + tn + r;     // cols tn..tn+15
  const float* B1 = B0 + 16;                             // cols tn+16..tn+31

  v8f z = {0.f,0.f,0.f,0.f,0.f,0.f,0.f,0.f};
  v8f acc00 = z, acc01 = z, acc10 = z, acc11 = z;

#pragma unroll
  for (int k = 0; k < 256; k += 4) {
    v2f a0 = *(const v2f*)(A0 + k);          // global_load_b64, imm offset
    v2f a1 = *(const v2f*)(A1 + k);
    v2f b0, b1;
    b0.x = B0[k * 256];  b0.y = B0[k * 256 + 256];
    b1.x = B1[k * 256];  b1.y = B1[k * 256 + 256];
    acc00 = wmma4(a0, b0, acc00);
    acc01 = wmma4(a0, b1, acc01);
    acc10 = wmma4(a1, b0, acc10);
    acc11 = wmma4(a1, b1, acc11);
  }

  float alpha = alphaC;
  if (aP0) alpha *= *aP0;
  if (aP1) alpha *= *aP1;

  auto st = [&](v8f acc, int rowBase, int colBase) {
#pragma unroll
    for (int v = 0; v < 8; v++) {
      const int row = rowBase + v + (hi ? 8 : 0);
      const size_t off = (size_t)row * 256 + colBase + r;
      float val = alpha * acc[v];
      if (accumulate) val += Ci[off];
      Ci[off] = val;
    }
  };
  st(acc00, tm,      tn);
  st(acc01, tm,      tn + 16);
  st(acc10, tm + 16, tn);
  st(acc11, tm + 16, tn + 16);
}

// ---------------------------------------------------------------------------
// Generic batched GEMM (used only for x_low = P*y : M=64, N=65536, K=64).
// block = 256 threads = 8 waves arranged 2(M) x 4(N); each wave -> 16x16 tile.
// grid = (ceil(N/64), ceil(M/32), nimg)
// ---------------------------------------------------------------------------
__global__ void k_gemm(const float* __restrict__ A, int lda, long long sA,
                       const float* __restrict__ B, int ldb, long long sB,
                       float* __restrict__ C, int ldc, long long sC,
                       int M, int N, int K,
                       float alphaC, const float* aP0, const float* aP1,
                       int accumulate)
{
  const int img  = blockIdx.z;
  const float* Ai = A + (size_t)img * (size_t)sA;
  const float* Bi = B + (size_t)img * (size_t)sB;
  float*       Ci = C + (size_t)img * (size_t)sC;

  const int wave = threadIdx.x >> 5;
  const int lane = threadIdx.x & 31;
  const int wm = wave >> 2, wn = wave & 3;
  const int tm = blockIdx.y * 32 + wm * 16;
  const int tn = blockIdx.x * 64 + wn * 16;
  if (tm >= M || tn >= N) return;          // wave-uniform: EXEC stays all-ones

  const int r  = lane & 15;
  const int hi = lane >> 4;

  v8f acc = {0.f,0.f,0.f,0.f,0.f,0.f,0.f,0.f};
  const float* Arow = Ai + (size_t)(tm + r) * (size_t)lda;

#pragma unroll 4
  for (int k = 0; k < K; k += 4) {
    const int kb = k + (hi ? 2 : 0);
    v2f a, b;
    a.x = Arow[kb];
    a.y = Arow[kb + 1];
    b.x = Bi[(size_t)kb       * (size_t)ldb + tn + r];
    b.y = Bi[(size_t)(kb + 1) * (size_t)ldb + tn + r];
    acc = wmma4(a, b, acc);
  }

  float alpha = alphaC;
  if (aP0) alpha *= *aP0;
  if (aP1) alpha *= *aP1;

#pragma unroll
  for (int v = 0; v < 8; v++) {
    const int row = tm + v + (hi ? 8 : 0);
    const int col = tn + r;
    const size_t off = (size_t)row * (size_t)ldc + col;
    float val = alpha * acc[v];
    if (accumulate) val += Ci[off];
    Ci[off] = val;
  }
}

// ---------------------------------------------------------------------------
// Gram: G[b] = Y_b * Y_b^T, Y_b = 64 x 65536. Split-K (64 chunks of 1024),
// block = 128 threads (4 waves, one 16-row strip each), atomicAdd into G.
// ---------------------------------------------------------------------------
__global__ void k_gram(const float* __restrict__ y, float* __restrict__ G)
{
  const int b  = blockIdx.z;
  const int kc = blockIdx.x;
  const float* Y = y + (size_t)b * CCH * HW;
  const int wave = threadIdx.x >> 5, lane = threadIdx.x & 31;
  const int r = lane & 15, hi = lane >> 4;
  const int tm = wave * 16;

  v8f acc[4];
#pragma unroll
  for (int t = 0; t < 4; t++) acc[t] = (v8f){0.f,0.f,0.f,0.f,0.f,0.f,0.f,0.f};

  const int k0 = kc * 1024, k1 = k0 + 1024;
  for (int k = k0; k < k1; k += 4) {
    const int kb = k + (hi ? 2 : 0);
    v2f a;
    a.x = Y[(size_t)(tm + r) * HW + kb];
    a.y = Y[(size_t)(tm + r) * HW + kb + 1];
#pragma unroll
    for (int nt = 0; nt < 4; nt++) {
      v2f bb;                              // B[k][n] = Y[n][k] (transposed read)
      bb.x = Y[(size_t)(nt * 16 + r) * HW + kb];
      bb.y = Y[(size_t)(nt * 16 + r) * HW + kb + 1];
      acc[nt] = wmma4(a, bb, acc[nt]);
    }
  }
  float* Gb = G + (size_t)b * CCH * CCH;
#pragma unroll
  for (int nt = 0; nt < 4; nt++)
#pragma unroll
    for (int v = 0; v < 8; v++) {
      const int row = tm + v + (hi ? 8 : 0);
      const int col = nt * 16 + r;
      atomicAdd(&Gb[row * CCH + col], acc[nt][v]);
    }
}

// ---------------------------------------------------------------------------
// Parallel Jacobi eigensolver (64x64 symmetric) + rank-16 projector P=UrUr^T.
// One block (256 threads) per batch. Round-robin tournament ordering:
// 32 disjoint rotations per step, 63 steps per sweep.
// ---------------------------------------------------------------------------
__global__ void k_eigenP(const float* __restrict__ Gin, float* __restrict__ P)
{
  __shared__ float G[64][65];
  __shared__ float V[64][65];
  __shared__ float rc[32], rs[32];
  __shared__ int top[32], bot[32];
  __shared__ float eval[64];
  __shared__ int   sel[16];

  const int b = blockIdx.x;
  const int t = threadIdx.x;
  const float* Gb = Gin + (size_t)b * CCH * CCH;

  for (int idx = t; idx < 64 * 64; idx += 256) {
    int i = idx >> 6, j = idx & 63;
    G[i][j] = Gb[idx];
    V[i][j] = (i == j) ? 1.f : 0.f;
  }
  if (t < 32) { top[t] = 2 * t; bot[t] = 2 * t + 1; }
  __syncthreads();

  for (int sweep = 0; sweep < 10; sweep++) {
    for (int step = 0; step < 63; step++) {
      if (t < 32) {
        int p = top[t], q = bot[t]; if (p > q) { int z = p; p = q; q = z; }
        float apq = G[p][q];
        float c = 1.f, s = 0.f;
        if (fabsf(apq) > 1e-12f) {
          float tau = (G[q][q] - G[p][p]) / (2.f * apq);
          float tt  = ((tau >= 0.f) ? 1.f : -1.f) /
                      (fabsf(tau) + sqrtf(1.f + tau * tau));
          c = 1.f / sqrtf(1.f + tt * tt);
          s = tt * c;
        }
        rc[t] = c; rs[t] = s;
      }
      __syncthreads();
      // rows: (J^T G)
      for (int idx = t; idx < 32 * 64; idx += 256) {
        int pi = idx >> 6, j = idx & 63;
        int p = top[pi], q = bot[pi]; if (p > q) { int z = p; p = q; q = z; }
        float c = rc[pi], s = rs[pi];
        float gp = G[p][j], gq = G[q][j];
        G[p][j] = c * gp - s * gq;
        G[q][j] = s * gp + c * gq;
      }
      __syncthreads();
      // cols: (... J), and V = V J
      for (int idx = t; idx < 32 * 64; idx += 256) {
        int pi = idx >> 6, i = idx & 63;
        int p = top[pi], q = bot[pi]; if (p > q) { int z = p; p = q; q = z; }
        float c = rc[pi], s = rs[pi];
        float gp = G[i][p], gq = G[i][q];
        G[i][p] = c * gp - s * gq;
        G[i][q] = s * gp + c * gq;
        float vp = V[i][p], vq = V[i][q];
        V[i][p] = c * vp - s * vq;
        V[i][q] = s * vp + c * vq;
      }
      __syncthreads();
      if (t == 0) {                        // rotate tournament (top[0] fixed)
        int nt_[32], nb_[32];
        nt_[0] = top[0]; nt_[1] = bot[0];
        for (int i = 2; i < 32; i++) nt_[i] = top[i - 1];
        for (int i = 0; i < 31; i++) nb_[i] = bot[i + 1];
        nb_[31] = top[31];
        for (int i = 0; i < 32; i++) { top[i] = nt_[i]; bot[i] = nb_[i]; }
      }
      __syncthreads();
    }
  }

  if (t < 64) eval[t] = G[t][t];
  __syncthreads();
  if (t < 64) {                            // descending rank, ties by index
    float e = eval[t]; int rk = 0;
    for (int j = 0; j < 64; j++) {
      float ej = eval[j];
      if (ej > e || (ej == e && j < t)) rk++;
    }
    if (rk < 16) sel[rk] = t;
  }
  __syncthreads();
  float* Pb = P + (size_t)b * CCH * CCH;
  for (int idx = t; idx < 64 * 64; idx += 256) {
    int i = idx >> 6, j = idx & 63;
    float a = 0.f;
#pragma unroll
    for (int k = 0; k < 16; k++) { int col = sel[k]; a += V[i][col] * V[j][col]; }
    Pb[idx] = a;
  }
}

// ---------------------------------------------------------------------------
// Constant matrices: DCT D/D^T, DFT cos/sin (symmetric), hp mask M3, radial RG
// ---------------------------------------------------------------------------
__global__ void k_consts(float* Dm, float* DmT, float* Cm, float* Sm,
                         float* M3, float* RG)
{
  const int idx = blockIdx.x * blockDim.x + threadIdx.x;
  if (idx >= NPIX * NPIX) return;
  const int i = idx >> 8, j = idx & 255;
  const float PI = 3.14159265358979323846f;

  float sc = (i == 0) ? sqrtf(1.f / 256.f) : sqrtf(2.f / 256.f);
  int m2 = ((2 * j + 1) * i) % 1024;                 // arg reduction
  float d = sc * cosf(PI * (float)m2 / 512.f);
  Dm[idx] = d;
  DmT[j * 256 + i] = d;

  int m = (i * j) & 255;
  float ang = PI * (float)m / 128.f;                 // 2*pi*(i*j)/256
  Cm[idx] = cosf(ang);
  Sm[idx] = sinf(ang);

  float fi = (float)((i < 128) ? i : i - 256) / 256.f;
  float fj = (float)((j < 128) ? j : j - 256) / 256.f;
  M3[idx] = (sqrtf(fi * fi + fj * fj) >= 0.2f) ? 1.f : 0.f;

  float yy = -1.f + 2.f * (float)i / 255.f;
  float xx = -1.f + 2.f * (float)j / 255.f;
  RG[idx] = sqrtf(xx * xx + yy * yy);
}

__device__ __forceinline__ float bw_gain(int i, int j, float fc, float n)
{
  float yy = -1.f + 2.f * (float)i / 255.f;
  float xx = -1.f + 2.f * (float)j / 255.f;
  float r = sqrtf(xx * xx + yy * yy);
  return 1.f / sqrtf(1.f + powf(r / fc, 2.f * n));
}

// Butterworth gain, Hermitian-extended to the full spectrum (irfft2-equivalent)
__global__ void k_m4(float* M4, const float* log_fc, const float* log_n)
{
  const int idx = blockIdx.x * blockDim.x + threadIdx.x;
  if (idx >= NPIX * NPIX) return;
  const int i = idx >> 8, j = idx & 255;
  const float fc = expf(*log_fc) + 1e-4f;
  const float n  = expf(*log_n);
  M4[idx] = (j <= 128) ? bw_gain(i, j, fc, n)
                       : bw_gain((256 - i) & 255, 256 - j, fc, n);
}

// ------------------------- elementwise kernels -----------------------------
__global__ void k_mask_dct(float* T, const float* RG, const float* alpha)
{
  size_t gid = (size_t)blockIdx.x * blockDim.x + threadIdx.x;
  if (gid >= (size_t)NIMG * HW) return;
  int img = (int)(gid >> 16), pix = (int)(gid & 65535);
  float a = alpha[img & 63];
  T[gid] *= (1.f - a) + a * RG[pix];
}

__global__ void k_mask2(float* Zr, float* Zi, const float* M)
{
  size_t gid = (size_t)blockIdx.x * blockDim.x + threadIdx.x;
  if (gid >= (size_t)NIMG * HW) return;
  float m = M[(int)(gid & 65535)];
  Zr[gid] *= m;
  Zi[gid] *= m;
}

__device__ __forceinline__ float shrinkf(float t, float l)
{
  return copysignf(fmaxf(fabsf(t) - l, 0.f), t);
}

__global__ void k_wav(float* y, const float* lam, const float* gw)
{
  size_t gid = (size_t)blockIdx.x * blockDim.x + threadIdx.x;
  if (gid >= (size_t)NIMG * 128 * 128) return;
  int img = (int)(gid >> 14), blk = (int)(gid & 16383);
  int bi = blk >> 7, bj = blk & 127;
  float l = lam[img & 63];
  float g = *gw;
  size_t base = (size_t)img * HW + (size_t)(2 * bi) * 256 + (size_t)(2 * bj);
  float a = y[base], b = y[base + 1], c = y[base + 256], d = y[base + 257];
  float ll = .5f * (a + b + c + d);
  float lh = shrinkf(.5f * (a + b - c - d), l);
  float hl = shrinkf(.5f * (a - b + c - d), l);
  float hh = shrinkf(.5f * (a - b - c + d), l);
  y[base]       = a + g * .5f * (ll + lh + hl + hh);
  y[base + 1]   = b + g * .5f * (ll + lh - hl - hh);
  y[base + 256] = c + g * .5f * (ll - lh + hl - hh);
  y[base + 257] = d + g * .5f * (ll - lh - hl + hh);
}

__global__ void k_zero(float* p, long long n)
{
  size_t gid = (size_t)blockIdx.x * blockDim.x + threadIdx.x;
  if (gid < (size_t)n) p[gid] = 0.f;
}

__global__ void k_axpy(float* y, const float* s, const float* g)
{
  size_t gid = (size_t)blockIdx.x * blockDim.x + threadIdx.x;
  if (gid >= (size_t)NIMG * HW) return;
  y[gid] += (*g) * s[gid];
}

// mean |rfft2(y, ortho)| per image: one block per image, reduce 256x129 entries
__global__ void k_amp(const float* Zr, const float* Zi, float* v)
{
  __shared__ float red[256];
  const int img = blockIdx.x;
  const float* zr = Zr + (size_t)img * HW;
  const float* zi = Zi + (size_t)img * HW;
  float s = 0.f;
  for (int idx = threadIdx.x; idx < 256 * 129; idx += 256) {
    int i = idx / 129, j = idx - i * 129;
    size_t o = (size_t)i * 256 + j;
    float a = zr[o], b = zi[o];
    s += sqrtf(a * a + b * b);
  }
  red[threadIdx.x] = s;
  __syncthreads();
  for (int st = 128; st > 0; st >>= 1) {
    if (threadIdx.x < st) red[threadIdx.x] += red[threadIdx.x + st];
    __syncthreads();
  }
  if (threadIdx.x == 0)
    v[img] = red[0] / (256.f * 256.f * 129.f);   // /256 ortho, /(256*129) mean
}

__global__ void k_mlp(const float* v, const float* w1, const float* w2, float* w)
{
  __shared__ float h[BB][4];
  const int t = threadIdx.x;
  if (t < 32) {
    int b = t >> 2, k = t & 3;
    float acc = 0.f;
    for (int c = 0; c < 64; c++) acc += v[b * 64 + c] * w1[k * 64 + c];
    h[b][k] = fmaxf(acc, 0.f);
  }
  __syncthreads();
  for (int idx = t; idx < BB * CCH; idx += 256) {
    int b = idx >> 6, c = idx & 63;
    float acc = 0.f;
#pragma unroll
    for (int k = 0; k < 4; k++) acc += h[b][k] * w2[c * 4 + k];
    w[idx] = 1.f / (1.f + expf(-acc));
  }
}

__global__ void k_final(float* y, const float* w)
{
  size_t gid = (size_t)blockIdx.x * blockDim.x + threadIdx.x;
  if (gid >= (size_t)NIMG * HW) return;
  y[gid] *= (1.f + w[(int)(gid >> 16)]);
}

// ---------------------------------------------------------------------------
// Host side
// ---------------------------------------------------------------------------
static void gemm(hipStream_t s,
                 const float* A, int lda, long long sA,
                 const float* B, int ldb, long long sB,
                 float* C, int ldc, long long sC,
                 int M, int N, int K, float alphaC,
                 const float* p0, const float* p1, int accum, int nimg)
{
  if (M == 256 && N == 256 && K == 256 &&
      lda == 256 && ldb == 256 && ldc == 256) {
    dim3 grid(2, 4, nimg);                     // 128(N) x 64(M) block tiles
    k_gemm256<<<grid, 256, 0, s>>>(A, sA, B, sB, C, sC,
                                   alphaC, p0, p1, accum);
  } else {
    dim3 grid((N + 63) / 64, (M + 31) / 32, nimg);
    k_gemm<<<grid, 256, 0, s>>>(A, lda, sA, B, ldb, sB, C, ldc, sC,
                                M, N, K, alphaC, p0, p1, accum);
  }
}

extern "C" void kernel_launch(void* const* d_in, const int* in_sizes, int n_in,
                              void* d_out, int out_size, void* d_ws, size_t ws_size,
                              hipStream_t stream)
{
  const float* x      = (const float*)d_in[0];
  const float* alpha  = (const float*)d_in[1];
  const float* lam    = (const float*)d_in[2];
  const float* scale  = (const float*)d_in[3];
  const float* log_fc = (const float*)d_in[4];
  const float* log_n  = (const float*)d_in[5];
  const float* g_dct  = (const float*)d_in[6];
  const float* g_wav  = (const float*)d_in[7];
  const float* g_lr   = (const float*)d_in[8];
  const float* g_bw   = (const float*)d_in[9];
  const float* w1     = (const float*)d_in[10];
  const float* w2     = (const float*)d_in[11];
  float* y = (float*)d_out;

  const size_t TEN = (size_t)NIMG * HW;
  float* ws   = (float*)d_ws;
  float* buf0 = ws;
  float* buf1 = buf0 + TEN;
  float* buf2 = buf1 + TEN;
  float* buf3 = buf2 + TEN;
  float* Dm   = buf3 + TEN;
  float* DmT  = Dm  + HW;
  float* Cm   = DmT + HW;
  float* Sm   = Cm  + HW;
  float* M3   = Sm  + HW;
  float* M4   = M3  + HW;
  float* RG   = M4  + HW;
  float* Gg   = RG  + HW;
  float* Pp   = Gg  + (size_t)BB * CCH * CCH;
  float* vv   = Pp  + (size_t)BB * CCH * CCH;
  float* ww   = vv  + 512;
  const size_t need = (size_t)((ww + 512) - ws) * sizeof(float);
  if (ws_size < need) return;                    // cannot run without scratch

  const int EB = (int)((TEN + 255) / 256);       // elementwise grid
  const float INVN2 = 1.f / 65536.f;

  k_consts<<<256, 256, 0, stream>>>(Dm, DmT, Cm, Sm, M3, RG);
  k_m4<<<256, 256, 0, stream>>>(M4, log_fc, log_n);
  hipMemcpyAsync(y, x, TEN * sizeof(float), hipMemcpyDeviceToDevice, stream);

  // ---- Stage 1: y += gamma_dct * D^T (mask o (D x D^T)) D --------------
  gemm(stream, Dm,256,0,     x,256,HW,   buf0,256,HW, 256,256,256, 1.f, 0,0, 0, NIMG);
  gemm(stream, buf0,256,HW,  DmT,256,0,  buf1,256,HW, 256,256,256, 1.f, 0,0, 0, NIMG);
  k_mask_dct<<<EB, 256, 0, stream>>>(buf1, RG, alpha);
  gemm(stream, DmT,256,0,    buf1,256,HW, buf0,256,HW, 256,256,256, 1.f, 0,0, 0, NIMG);
  gemm(stream, buf0,256,HW,  Dm,256,0,   y,256,HW,   256,256,256, 1.f, g_dct,0, 1, NIMG);

  // ---- Stage 2: Haar shrink (in place, one thread per 2x2 block) -------
  k_wav<<<(int)((TEN / 4 + 255) / 256), 256, 0, stream>>>(y, lam, g_wav);

  // ---- Stage 3: low-rank (Gram + Jacobi + projector) + FFT high-pass ---
  k_zero<<<(BB * CCH * CCH + 255) / 256, 256, 0, stream>>>(Gg, BB * CCH * CCH);
  k_gram<<<dim3(64, 1, BB), 128, 0, stream>>>(y, Gg);
  k_eigenP<<<BB, 256, 0, stream>>>(Gg, Pp);
  // forward FFT: rows then cols (C,S symmetric => right-multiply by C/S)
  gemm(stream, y,256,HW,   Cm,256,0, buf0,256,HW, 256,256,256,  1.f, 0,0, 0, NIMG); // A1
  gemm(stream, y,256,HW,   Sm,256,0, buf1,256,HW, 256,256,256, -1.f, 0,0, 0, NIMG); // B1
  gemm(stream, Cm,256,0, buf0,256,HW, buf2,256,HW, 256,256,256,  1.f, 0,0, 0, NIMG); // Zr
  gemm(stream, Sm,256,0, buf1,256,HW, buf2,256,HW, 256,256,256,  1.f, 0,0, 1, NIMG);
  gemm(stream, Cm,256,0, buf1,256,HW, buf3,256,HW, 256,256,256,  1.f, 0,0, 0, NIMG); // Zi
  gemm(stream, Sm,256,0, buf0,256,HW, buf3,256,HW, 256,256,256, -1.f, 0,0, 1, NIMG);
  k_mask2<<<EB, 256, 0, stream>>>(buf2, buf3, M3);
  gemm(stream, Cm,256,0, buf2,256,HW, buf0,256,HW, 256,256,256,  1.f, 0,0, 0, NIMG); // Wr
  gemm(stream, Sm,256,0, buf3,256,HW, buf0,256,HW, 256,256,256, -1.f, 0,0, 1, NIMG);
  gemm(stream, Cm,256,0, buf3,256,HW, buf1,256,HW, 256,256,256,  1.f, 0,0, 0, NIMG); // Wi
  gemm(stream, Sm,256,0, buf2,256,HW, buf1,256,HW, 256,256,256,  1.f, 0,0, 1, NIMG);
  // x_low = P*y (reads y BEFORE y is modified below; additions commute)
  gemm(stream, Pp,64,(long long)CCH*CCH, y,HW,(long long)CCH*HW,
       buf2,HW,(long long)CCH*HW, 64,HW,64, 1.f, 0,0, 0, BB);
  k_axpy<<<EB, 256, 0, stream>>>(y, buf2, g_lr);
  gemm(stream, buf0,256,HW, Cm,256,0, y,256,HW, 256,256,256,  INVN2, g_lr,scale, 1, NIMG);
  gemm(stream, buf1,256,HW, Sm,256,0, y,256,HW, 256,256,256, -INVN2, g_lr,scale, 1, NIMG);

  // ---- Stage 4: Butterworth = same masked-filter pipeline with M4 ------
  gemm(stream, y,256,HW,   Cm,256,0, buf0,256,HW, 256,256,256,  1.f, 0,0, 0, NIMG);
  gemm(stream, y,256,HW,   Sm,256,0, buf1,256,HW, 256,256,256, -1.f, 0,0, 0, NIMG);
  gemm(stream, Cm,256,0, buf0,256,HW, buf2,256,HW, 256,256,256,  1.f, 0,0, 0, NIMG);
  gemm(stream, Sm,256,0, buf1,256,HW, buf2,256,HW, 256,256,256,  1.f, 0,0, 1, NIMG);
  gemm(stream, Cm,256,0, buf1,256,HW, buf3,256,HW, 256,256,256,  1.f, 0,0, 0, NIMG);
  gemm(stream, Sm,256,0, buf0,256,HW, buf3,256,HW, 256,256,256, -1.f, 0,0, 1, NIMG);
  k_mask2<<<EB, 256, 0, stream>>>(buf2, buf3, M4);
  gemm(stream, Cm,256,0, buf2,256,HW, buf0,256,HW, 256,256,256,  1.f, 0,0, 0, NIMG);
  gemm(stream, Sm,256,0, buf3,256,HW, buf0,256,HW, 256,256,256, -1.f, 0,0, 1, NIMG);
  gemm(stream, Cm,256,0, buf3,256,HW, buf1,256,HW, 256,256,256,  1.f, 0,0, 0, NIMG);
  gemm(stream, Sm,256,0, buf2,256,HW, buf1,256,HW, 256,256,256,  1.f, 0,0, 1, NIMG);
  gemm(stream, buf0,256,HW, Cm,256,0, y,256,HW, 256,256,256,  INVN2, g_bw,0, 1, NIMG);
  gemm(stream, buf1,256,HW, Sm,256,0, y,256,HW, 256,256,256, -INVN2, g_bw,0, 1, NIMG);

  // ---- Stage 5: SFU collapses to y *= (1 + w); w from rfft2 amp means --
  gemm(stream, y,256,HW,   Cm,256,0, buf0,256,HW, 256,256,256,  1.f, 0,0, 0, NIMG);
  gemm(stream, y,256,HW,   Sm,256,0, buf1,256,HW, 256,256,256, -1.f, 0,0, 0, NIMG);
  gemm(stream, Cm,256,0, buf0,256,HW, buf2,256,HW, 256,256,256,  1.f, 0,0, 0, NIMG);
  gemm(stream, Sm,256,0, buf1,256,HW, buf2,256,HW, 256,256,256,  1.f, 0,0, 1, NIMG);
  gemm(stream, Cm,256,0, buf1,256,HW, buf3,256,HW, 256,256,256,  1.f, 0,0, 0, NIMG);
  gemm(stream, Sm,256,0, buf0,256,HW, buf3,256,HW, 256,256,256, -1.f, 0,0, 1, NIMG);
  k_amp<<<NIMG, 256, 0, stream>>>(buf2, buf3, vv);
  k_mlp<<<1, 256, 0, stream>>>(vv, w1, w2, ww);
  k_final<<<EB, 256, 0, stream>>>(y, ww);
}